// Sonata3DSegLevelEncoder_38646115729987
// MI455X (gfx1250) — compile-verified
//
#include <hip/hip_runtime.h>
#include <hip/hip_bf16.h>

typedef __attribute__((ext_vector_type(2))) float v2f;
typedef __attribute__((ext_vector_type(8))) float v8f;

#define BATCH 8
#define PTS   65536
#define NPTS  (BATCH * PTS)          // 524288
#define MAXSEG 512
#define HID    256
#define CH_TOT 512                   // 64 + 64 + 128 + 256 concatenated
#define CHUNK  4096                  // points per scatter block
#define NCHUNK (PTS / CHUNK)         // 16

// ---------------------------------------------------------------------------
// Workspace layout (bytes):
//   idx2   : NPTS * 4                      @ 0
//   idx3   : NPTS * 4                      @ 2097152
//   sums   : BATCH*MAXSEG*CH_TOT * 4       @ 4194304   (8 MB)
//   counts : BATCH*MAXSEG * 4 (uint)       @ 8388608
//   recip  : BATCH*MAXSEG * 4 (float)      @ 8404992
// ---------------------------------------------------------------------------
#define WS_IDX2   0
#define WS_IDX3   2097152
#define WS_SUMS   4194304
#define WS_COUNTS 8388608
#define WS_RECIP  8404992

// ---- Kernel 1: compose inverse index chains to finest resolution ----------
__global__ void sonata_compose_idx(const int* __restrict__ inv1,
                                   const int* __restrict__ inv2,
                                   const int* __restrict__ inv3,
                                   int* __restrict__ idx2,
                                   int* __restrict__ idx3) {
  int i = blockIdx.x * blockDim.x + threadIdx.x;
  if (i < NPTS) {
    int j1 = inv1[i];
    int j2 = inv2[j1];
    idx2[i] = j2;
    idx3[i] = inv3[j2];
  }
}

// ---- Kernel 2: per-(batch,segment) point counts (LDS histogram) -----------
__global__ void sonata_count(const int* __restrict__ p2s,
                             unsigned* __restrict__ counts) {
  __shared__ unsigned hist[MAXSEG];
  int b = blockIdx.y;
  int pbase = blockIdx.x * CHUNK;
  for (int i = threadIdx.x; i < MAXSEG; i += blockDim.x) hist[i] = 0u;
  __syncthreads();
  const int* row = p2s + (long)b * PTS;
  for (int p = pbase + threadIdx.x; p < pbase + CHUNK; p += blockDim.x)
    atomicAdd(&hist[row[p]], 1u);
  __syncthreads();
  for (int i = threadIdx.x; i < MAXSEG; i += blockDim.x)
    if (hist[i]) atomicAdd(&counts[b * MAXSEG + i], hist[i]);
}

// ---- Kernel 3: reciprocal of counts (mean divisor, empty seg -> /1) -------
__global__ void sonata_recip(const unsigned* __restrict__ counts,
                             float* __restrict__ recip) {
  int i = blockIdx.x * blockDim.x + threadIdx.x;
  if (i < BATCH * MAXSEG) {
    unsigned c = counts[i];
    recip[i] = 1.0f / (float)(c > 0u ? c : 1u);
  }
}

// ---- Kernel 4: gather + scatter-sum, 64-channel groups, LDS accumulators --
// grid = (NCHUNK, 8 channel-groups, BATCH), 256 threads, 128 KB dynamic LDS.
// Channel map: g=0 -> feat0[0:64]; g=1 -> feat1[inv1][0:64];
//              g=2,3 -> feat2[idx2][(g-2)*64 : +64];
//              g=4..7 -> feat3[idx3][(g-4)*64 : +64].
__global__ void sonata_scatter(const float* __restrict__ feat0,
                               const float* __restrict__ feat1,
                               const float* __restrict__ feat2,
                               const float* __restrict__ feat3,
                               const int* __restrict__ inv1,
                               const int* __restrict__ idx2,
                               const int* __restrict__ idx3,
                               const int* __restrict__ p2s,
                               float* __restrict__ sums) {
  extern __shared__ float acc[];          // MAXSEG * 64 floats = 128 KB
  const int b = blockIdx.z;
  const int g = blockIdx.y;
  const int pbase = blockIdx.x * CHUNK;
  for (int i = threadIdx.x; i < MAXSEG * 64; i += blockDim.x) acc[i] = 0.0f;
  __syncthreads();

  const int tch = threadIdx.x & 63;       // channel within group
  const int tp  = threadIdx.x >> 6;       // 4 points in flight
  for (int pp = tp; pp < CHUNK; pp += 4) {
    const long gi = (long)b * PTS + (pbase + pp);
    const int seg = p2s[gi];
    float v;
    if (g == 0)      { v = feat0[gi * 64 + tch]; }
    else if (g == 1) { int j = inv1[gi]; v = feat1[(long)j * 64 + tch]; }
    else if (g < 4)  { int j = idx2[gi]; v = feat2[(long)j * 128 + (g - 2) * 64 + tch]; }
    else             { int j = idx3[gi]; v = feat3[(long)j * 256 + (g - 4) * 64 + tch]; }
    atomicAdd(&acc[seg * 64 + tch], v);   // ds_add_f32
  }
  __syncthreads();

  float* dst = sums + (long)b * MAXSEG * CH_TOT + g * 64;
  for (int i = threadIdx.x; i < MAXSEG * 64; i += blockDim.x) {
    float v = acc[i];
    if (v != 0.0f) atomicAdd(&dst[(long)(i >> 6) * CH_TOT + (i & 63)], v);
  }
}

// ---- Kernel 5: mean -> Linear (WMMA f32 16x16x4) -> bias -> LayerNorm -----
// grid = BATCH*MAXSEG/16 = 256 blocks, 512 threads (16 waves, wave32).
// Each block owns 16 rows (segments) x 256 outputs; wave w owns N-tile w.
__global__ __launch_bounds__(512)
void sonata_gemm_ln(const float* __restrict__ sums,
                    const float* __restrict__ recip,
                    const float* __restrict__ W,     // [K, HID] row-major
                    const float* __restrict__ bias,  // [HID]
                    const float* __restrict__ gamma, // [HID]
                    const float* __restrict__ beta,  // [HID]
                    float* __restrict__ out,         // scale slot base
                    int K, int ch_off) {
  __shared__ float Alds[16 * 257];   // 16 rows x K (padded stride K+1), K<=256
  __shared__ float hbuf[16 * HID];   // post-GEMM tile for LayerNorm

  const int R = blockIdx.x * 16;     // global row base (b*512 + seg)
  const int tid = threadIdx.x;

  // Load A tile = segment means (sums * recip), padded stride K+1.
  for (int i = tid; i < 16 * K; i += blockDim.x) {
    int row = i / K, k = i - row * K;
    float m = sums[(long)(R + row) * CH_TOT + ch_off + k] * recip[R + row];
    Alds[row * (K + 1) + k] = m;
  }
  __syncthreads();

  const int wave = tid >> 5;
  const int lane = tid & 31;
  const int half = lane >> 4;        // K-pair selector per §7.12.2
  const int ml   = lane & 15;        // M for A-frag, N for B-frag
  const int n0   = wave * 16;

  v8f c = {};
  for (int k0 = 0; k0 < K; k0 += 4) {
    v2f a, bfrag;
    const float* arow = &Alds[ml * (K + 1) + k0 + 2 * half];
    a.x = arow[0];
    a.y = arow[1];
    const float* wp = &W[(long)(k0 + 2 * half) * HID + n0 + ml];
    bfrag.x = wp[0];
    bfrag.y = wp[HID];
    c = __builtin_amdgcn_wmma_f32_16x16x4_f32(false, a, false, bfrag,
                                              (short)0, c, false, false);
  }

  // Bias, deposit D tile into LDS (row = v + 8*half, col = n0 + ml).
  const float bv = bias[n0 + ml];
  #pragma unroll
  for (int v = 0; v < 8; ++v)
    hbuf[(v + 8 * half) * HID + n0 + ml] = c[v] + bv;
  __syncthreads();

  // LayerNorm: wave w normalizes row w (256 cols, 8 per lane).
  float x[8], s = 0.0f, q = 0.0f;
  #pragma unroll
  for (int j = 0; j < 8; ++j) {
    x[j] = hbuf[wave * HID + j * 32 + lane];
    s += x[j];
    q += x[j] * x[j];
  }
  #pragma unroll
  for (int off = 16; off >= 1; off >>= 1) {
    s += __shfl_xor(s, off, 32);
    q += __shfl_xor(q, off, 32);
  }
  const float mean = s * (1.0f / 256.0f);
  const float var  = q * (1.0f / 256.0f) - mean * mean;
  const float rstd = rsqrtf(var + 1e-5f);
  float* orow = out + (long)(R + wave) * HID;
  #pragma unroll
  for (int j = 0; j < 8; ++j) {
    int col = j * 32 + lane;
    orow[col] = (x[j] - mean) * rstd * gamma[col] + beta[col];
  }
}

// ---------------------------------------------------------------------------
extern "C" void kernel_launch(void* const* d_in, const int* in_sizes, int n_in,
                              void* d_out, int out_size, void* d_ws, size_t ws_size,
                              hipStream_t stream) {
  const float* feat0 = (const float*)d_in[0];
  const float* feat1 = (const float*)d_in[1];
  const float* feat2 = (const float*)d_in[2];
  const float* feat3 = (const float*)d_in[3];
  const int*   inv1  = (const int*)d_in[4];
  const int*   inv2  = (const int*)d_in[5];
  const int*   inv3  = (const int*)d_in[6];
  const int*   p2s   = (const int*)d_in[7];
  // W{i},b{i},g{i},bt{i} at 8+4i .. 11+4i
  const float* Ws[4]  = {(const float*)d_in[8],  (const float*)d_in[12],
                         (const float*)d_in[16], (const float*)d_in[20]};
  const float* bs[4]  = {(const float*)d_in[9],  (const float*)d_in[13],
                         (const float*)d_in[17], (const float*)d_in[21]};
  const float* gs[4]  = {(const float*)d_in[10], (const float*)d_in[14],
                         (const float*)d_in[18], (const float*)d_in[22]};
  const float* bts[4] = {(const float*)d_in[11], (const float*)d_in[15],
                         (const float*)d_in[19], (const float*)d_in[23]};
  float* out = (float*)d_out;

  char* ws = (char*)d_ws;
  int*      idx2   = (int*)(ws + WS_IDX2);
  int*      idx3   = (int*)(ws + WS_IDX3);
  float*    sums   = (float*)(ws + WS_SUMS);
  unsigned* counts = (unsigned*)(ws + WS_COUNTS);
  float*    recip  = (float*)(ws + WS_RECIP);

  // Zero accumulators (sums + counts are contiguous). Capturable memset node.
  hipMemsetAsync(ws + WS_SUMS,
                 0, (size_t)BATCH * MAXSEG * CH_TOT * 4 + (size_t)BATCH * MAXSEG * 4,
                 stream);

  sonata_compose_idx<<<NPTS / 256, 256, 0, stream>>>(inv1, inv2, inv3, idx2, idx3);

  sonata_count<<<dim3(NCHUNK, BATCH), 256, 0, stream>>>(p2s, counts);
  sonata_recip<<<(BATCH * MAXSEG) / 256, 256, 0, stream>>>(counts, recip);

  sonata_scatter<<<dim3(NCHUNK, 8, BATCH), 256, MAXSEG * 64 * sizeof(float), stream>>>(
      feat0, feat1, feat2, feat3, inv1, idx2, idx3, p2s, sums);

  // Per-scale GEMM+LN. Output is coarsest-first: scale s -> slot (3 - s).
  const int  Kc[4]  = {64, 64, 128, 256};
  const int  off[4] = {0, 64, 128, 256};
  const long slot   = (long)BATCH * MAXSEG * HID;
  for (int s = 0; s < 4; ++s) {
    sonata_gemm_ln<<<(BATCH * MAXSEG) / 16, 512, 0, stream>>>(
        sums, recip, Ws[s], bs[s], gs[s], bts[s],
        out + (3 - s) * slot, Kc[s], off[s]);
  }
}